// ShiftNMF_23106924053335
// MI455X (gfx1250) — compile-verified
//
#include <hip/hip_runtime.h>
#include <cstdint>

typedef float v2f __attribute__((ext_vector_type(2)));
typedef float v8f __attribute__((ext_vector_type(8)));

constexpr int kRank  = 8;
constexpr int kNRows = 1024;
constexpr int kM     = 8192;
constexpr int kN1    = 64;    // outer radix (four-step FFT)
constexpr int kN2    = 128;   // inner radix
constexpr int kMTile = 512;   // stage-2 m-tile per block

static constexpr float kTwoPi = 6.28318530717958647692f;

static __device__ __forceinline__ v8f wmma_f32(v2f a, v2f b, v8f c) {
  // (neg_a, A, neg_b, B, c_mod, C, reuse_a, reuse_b)
  return __builtin_amdgcn_wmma_f32_16x16x4_f32(false, a, false, b, (short)0, c,
                                               false, false);
}

// -------------------------------------------------------------------------
// Stage 0: Xsp[r][n1*128 + n2] = softplus(H[r][n1 + 64*n2])
// Branchless softplus, written once, already in the stage-1 GEMM layout
// (row n1, contiguous in K = n2) so stage-1 A-fragments are b64 loads.
// -------------------------------------------------------------------------
__global__ __launch_bounds__(256) void shiftnmf_stage0_softplus(
    const float* __restrict__ H, float* __restrict__ Xsp) {
  const int idx = blockIdx.x * 256 + threadIdx.x;   // 0 .. 65535
  const int r   = idx >> 13;
  const int rem = idx & (kM - 1);
  const int n1  = rem >> 7;          // 0..63
  const int n2  = rem & (kN2 - 1);   // 0..127
  const float x = H[r * kM + n1 + kN1 * n2];
  // softplus(x) = max(x,0) + log1p(exp(-|x|))  (branchless, EXEC stays full)
  const float sp = fmaxf(x, 0.0f) + log1pf(__expf(-fabsf(x)));
  Xsp[idx] = sp;
}

// -------------------------------------------------------------------------
// Stage 1: Hf[r,:] = FFT_8192(softplus(H[r,:])) via four-step (64 x 128).
// Both radix stages are evaluated as f32 WMMA GEMMs; twiddles built with
// on-the-fly sincos (angles reduced mod 2pi exactly via integer masking).
// One block per rank, 128 threads = 4 waves; each wave owns 16x16 D tiles.
// -------------------------------------------------------------------------
__global__ __launch_bounds__(128) void shiftnmf_stage1_fft(
    const float* __restrict__ Xsp, float2* __restrict__ Hf) {
  const int r     = blockIdx.x;
  const int tid   = threadIdx.x;
  const int lane  = tid & 31;
  const int wave  = tid >> 5;
  const int lrow  = lane & 15;        // M/N index within 16x16 tile
  const int khalf = (lane >> 4) * 2;  // K offset for upper lane half

  __shared__ float yRe[kN1 * kN2];    // 32 KB: Y' real
  __shared__ float yIm[kN1 * kN2];    // 32 KB: Y' imag

  // ---- step A: Y[n1,k2] = sum_n2 Xsp[n1,n2] * e^{-2pi i n2 k2/128}
  // ---- step B: Y'[n1,k2] = Y[n1,k2] * e^{-2pi i n1 k2 / 8192}
  for (int t = 0; t < 8; ++t) {
    const int tile = wave * 8 + t;      // 32 tiles: 4 (n1) x 8 (k2)
    const int n1T  = (tile & 3) * 16;
    const int k2T  = (tile >> 2) * 16;
    v8f cRe = {0.f, 0.f, 0.f, 0.f, 0.f, 0.f, 0.f, 0.f};
    v8f cIm = {0.f, 0.f, 0.f, 0.f, 0.f, 0.f, 0.f, 0.f};
    const int nCol = k2T + lrow;
    const float* __restrict__ Arow = Xsp + r * kM + (n1T + lrow) * kN2;
    for (int k0 = 0; k0 < kN2; k0 += 4) {
      const int ka = k0 + khalf;
      v2f a;
      a.x = Arow[ka];        // contiguous pair -> global_load_b64
      a.y = Arow[ka + 1];
      float s0, c0, s1, c1;
      __sincosf(kTwoPi * (float)((ka * nCol) & (kN2 - 1)) * (1.0f / kN2), &s0, &c0);
      __sincosf(kTwoPi * (float)(((ka + 1) * nCol) & (kN2 - 1)) * (1.0f / kN2), &s1, &c1);
      v2f bC; bC.x = c0;  bC.y = c1;   // Re(D128)
      v2f bS; bS.x = -s0; bS.y = -s1;  // Im(D128)
      cRe = wmma_f32(a, bC, cRe);
      cIm = wmma_f32(a, bS, cIm);
    }
#pragma unroll
    for (int j = 0; j < 8; ++j) {
      const int n1 = n1T + j + 8 * (lane >> 4);
      const int k2 = k2T + lrow;
      float s, c;
      __sincosf(kTwoPi * (float)((n1 * k2) & (kM - 1)) * (1.0f / kM), &s, &c);
      const float re = cRe[j], im = cIm[j];
      yRe[n1 * kN2 + k2] = re * c + im * s;   // (re+i im)(c - i s)
      yIm[n1 * kN2 + k2] = im * c - re * s;
    }
  }
  __syncthreads();

  // ---- step C: Z[k1,k2] = sum_n1 e^{-2pi i n1 k1/64} * Y'[n1,k2]
  // Complex GEMM as 4 real WMMA accumulator chains.
  float2* __restrict__ HfRow = Hf + r * kM;
  for (int t = 0; t < 8; ++t) {
    const int tile = wave * 8 + t;
    const int k1T  = (tile & 3) * 16;
    const int k2T  = (tile >> 2) * 16;
    v8f g1 = {0.f, 0.f, 0.f, 0.f, 0.f, 0.f, 0.f, 0.f};
    v8f g2 = {0.f, 0.f, 0.f, 0.f, 0.f, 0.f, 0.f, 0.f};
    v8f g3 = {0.f, 0.f, 0.f, 0.f, 0.f, 0.f, 0.f, 0.f};
    v8f g4 = {0.f, 0.f, 0.f, 0.f, 0.f, 0.f, 0.f, 0.f};
    const int k1   = k1T + lrow;
    const int nCol = k2T + lrow;
    for (int k0 = 0; k0 < kN1; k0 += 4) {
      const int na = k0 + khalf;
      float s0, c0, s1, c1;
      __sincosf(kTwoPi * (float)((k1 * na) & (kN1 - 1)) * (1.0f / kN1), &s0, &c0);
      __sincosf(kTwoPi * (float)((k1 * (na + 1)) & (kN1 - 1)) * (1.0f / kN1), &s1, &c1);
      v2f aC; aC.x = c0; aC.y = c1;    // Re(D64)
      v2f aS; aS.x = s0; aS.y = s1;    // -Im(D64)
      v2f bRe; bRe.x = yRe[na * kN2 + nCol]; bRe.y = yRe[(na + 1) * kN2 + nCol];
      v2f bIm; bIm.x = yIm[na * kN2 + nCol]; bIm.y = yIm[(na + 1) * kN2 + nCol];
      g1 = wmma_f32(aC, bRe, g1);
      g2 = wmma_f32(aS, bIm, g2);
      g3 = wmma_f32(aC, bIm, g3);
      g4 = wmma_f32(aS, bRe, g4);
    }
#pragma unroll
    for (int j = 0; j < 8; ++j) {
      const int k1o = k1T + j + 8 * (lane >> 4);
      const int k2  = k2T + lrow;
      float2 z;
      z.x = g1[j] + g2[j];   // Zre = cos*bRe + sin*bIm
      z.y = g3[j] - g4[j];   // Zim = cos*bIm - sin*bRe
      HfRow[k1o * kN2 + k2] = z;   // k_out = 128*k1 + k2
    }
  }
}

// -------------------------------------------------------------------------
// Stage 2: V[n,m] = sum_r softmax(W)[n,r] * e^{-2pi i tau[n,r] m/M} * Hf[r,m]
// Block = 16 n-rows x 512 m-cols; Hf tile (32 KB) staged via async-to-LDS.
// -------------------------------------------------------------------------
__global__ __launch_bounds__(256) void shiftnmf_stage2(
    const float* __restrict__ W, const float* __restrict__ tau,
    const float2* __restrict__ Hf, float2* __restrict__ out) {
  const int mBase = blockIdx.x * kMTile;
  const int nBase = blockIdx.y * 16;
  const int tid   = threadIdx.x;

  __shared__ __align__(16) float2 hTile[kRank * kMTile];  // 32 KB
  __shared__ float swS[16 * kRank];
  __shared__ float tauS[16 * kRank];

  // Async DMA the 8 x 512-complex Hf tile into LDS (gfx1250 async path,
  // tracked on ASYNCcnt; 16 B per lane per row -> one op covers a row).
  {
    const unsigned ldsBase = (unsigned)(uintptr_t)(&hTile[0]);
    const unsigned vOff    = (unsigned)tid * 16u;
#pragma unroll
    for (int r = 0; r < kRank; ++r) {
      const unsigned long long src =
          (unsigned long long)(uintptr_t)(const void*)(Hf + r * kM + mBase);
      const unsigned lAddr = ldsBase + (unsigned)r * (unsigned)(kMTile * 8) + vOff;
      asm volatile("global_load_async_to_lds_b128 %0, %1, %2"
                   :: "v"(lAddr), "v"(vOff), "s"(src)
                   : "memory");
    }
  }

  // Row softmax of W (16 rows x 8) + tau staging; overlaps with async DMA.
  if (tid < 16) {
    const int n = nBase + tid;
    float w[kRank];
    float mx = -3.4e38f;
#pragma unroll
    for (int j = 0; j < kRank; ++j) {
      w[j] = W[n * kRank + j];
      mx = fmaxf(mx, w[j]);
    }
    float sum = 0.f;
#pragma unroll
    for (int j = 0; j < kRank; ++j) {
      w[j] = __expf(w[j] - mx);
      sum += w[j];
    }
    const float inv = 1.0f / sum;
#pragma unroll
    for (int j = 0; j < kRank; ++j) {
      swS[tid * kRank + j]  = w[j] * inv;
      tauS[tid * kRank + j] = tau[n * kRank + j];
    }
  }

  asm volatile("s_wait_asynccnt 0" ::: "memory");
  __syncthreads();

  // 16 rows x 512 cols = 8192 outputs; 32 per thread, coalesced in m.
  for (int it = 0; it < 32; ++it) {
    const int idx = it * 256 + tid;
    const int nl  = idx >> 9;             // /512: local n row
    const int ml  = idx & (kMTile - 1);   // local m col
    const int mg  = mBase + ml;
    const float fm = (float)mg * (1.0f / (float)kM);
    float re = 0.f, im = 0.f;
#pragma unroll
    for (int r = 0; r < kRank; ++r) {
      const float sw = swS[nl * kRank + r];
      const float th = kTwoPi * tauS[nl * kRank + r] * fm;  // |arg| < 2pi
      float s, c;
      __sincosf(th, &s, &c);
      const float2 hf = hTile[r * kMTile + ml];
      // (c - i s) * (hx + i hy), scaled by sw, accumulated
      re = __fmaf_rn(sw, __fmaf_rn(c, hf.x, s * hf.y), re);
      im = __fmaf_rn(sw, __fmaf_rn(c, hf.y, -s * hf.x), im);
    }
    float2 v;
    v.x = re;
    v.y = im;
    out[(nBase + nl) * kM + mg] = v;
  }
}

extern "C" void kernel_launch(void* const* d_in, const int* in_sizes, int n_in,
                              void* d_out, int out_size, void* d_ws,
                              size_t ws_size, hipStream_t stream) {
  (void)in_sizes; (void)n_in; (void)out_size; (void)ws_size;
  const float* W   = (const float*)d_in[0];   // (1024, 8)
  const float* H   = (const float*)d_in[1];   // (8, 8192)
  const float* tau = (const float*)d_in[2];   // (1024, 8)

  float2* Hf  = (float2*)d_ws;                   // 512 KB
  float*  Xsp = (float*)(Hf + kRank * kM);       // +256 KB, GEMM-layout softplus(H)
  float2* out = (float2*)d_out;                  // (1024, 8192) complex interleaved

  shiftnmf_stage0_softplus<<<(kRank * kM) / 256, 256, 0, stream>>>(H, Xsp);
  shiftnmf_stage1_fft<<<kRank, 128, 0, stream>>>(Xsp, Hf);
  shiftnmf_stage2<<<dim3(kM / kMTile, kNRows / 16), 256, 0, stream>>>(W, tau, Hf,
                                                                      out);
}